// PredictRetinaface_33208687132990
// MI455X (gfx1250) — compile-verified
//
#include <hip/hip_runtime.h>
#include <math.h>

// Problem constants (match reference)
#define B_    16
#define A_    200000
#define K_    300
#define CAP   8192          // per-batch candidate cap (~2.3% of 200k ≈ 4600 expected)
#define SORT_N 8192
#define THR_NMS 0.3f
#define NT    19            // ceil(K_/16)
#define PAD   (NT*16)       // 304

#define SCAN_ELEMS 2048
#define BLOCKS_PER_BATCH ((A_ + SCAN_ELEMS - 1) / SCAN_ELEMS)   // 98

typedef __attribute__((ext_vector_type(2))) float v2f;
typedef __attribute__((ext_vector_type(8))) float v8f;

// ---------------------------------------------------------------------------
// Kernel 1: reset per-batch candidate counters (workspace re-inited every
// launch for determinism).
// ---------------------------------------------------------------------------
__global__ void reset_counts_kernel(int* __restrict__ counts) {
  if (threadIdx.x < B_) counts[threadIdx.x] = 0;
}

// ---------------------------------------------------------------------------
// Kernel 2: stream p_conf (the only large mandatory read: 12.8 MB ≈ 0.55 us
// at 23.3 TB/s), compact survivors (logit >= 0 <=> sigmoid >= 0.5) into
// per-batch key lists. float4 loads cut VMEM issue count 4x.
// Key = (f32 score bits << 32) | ~anchor  -> descending u64 order equals
// descending score with smaller-index-first tie break (lax.top_k semantics).
// One global atomicAdd per block (LDS-staged) instead of per-element.
// ---------------------------------------------------------------------------
__global__ __launch_bounds__(256)
void scan_conf_kernel(const float* __restrict__ conf,
                      int* __restrict__ counts,
                      unsigned long long* __restrict__ keys) {
  __shared__ unsigned long long buf[SCAN_ELEMS];   // 16 KB
  __shared__ int lcnt;
  __shared__ int gbase;

  const int batch = blockIdx.x / BLOCKS_PER_BATCH;
  const int chunk = blockIdx.x % BLOCKS_PER_BATCH;
  const int a0    = chunk * SCAN_ELEMS;

  if (threadIdx.x == 0) lcnt = 0;
  __syncthreads();

  const float* cb = conf + (size_t)batch * A_;
  for (int q = threadIdx.x; q < (SCAN_ELEMS >> 2); q += blockDim.x) {
    const int a = a0 + q * 4;               // A_ % 4 == 0, chunks 16B aligned
    if (a + SCAN_ELEMS < A_) __builtin_prefetch(&cb[a + SCAN_ELEMS], 0, 1);
    if (a < A_) {
      const float4 x4 = *(const float4*)(cb + a);
      const float xs[4] = {x4.x, x4.y, x4.z, x4.w};
#pragma unroll
      for (int c = 0; c < 4; ++c) {
        const float x = xs[c];
        if (x >= 0.0f) {                    // sigmoid(x) >= 0.5
          const float s = 1.0f / (1.0f + __expf(-x));
          const unsigned long long key =
              ((unsigned long long)__float_as_uint(s) << 32) |
              (unsigned long long)(unsigned int)(~(a + c));
          const int p = atomicAdd(&lcnt, 1);   // < SCAN_ELEMS by construction
          buf[p] = key;
        }
      }
    }
  }
  __syncthreads();
  if (threadIdx.x == 0) gbase = atomicAdd(&counts[batch], lcnt);
  __syncthreads();
  const int n = lcnt, base = gbase;
  for (int t = threadIdx.x; t < n; t += blockDim.x) {
    const int pos = base + t;
    if (pos < CAP) keys[(size_t)batch * CAP + pos] = buf[t];
  }
}

// ---------------------------------------------------------------------------
// Kernel 3: per-batch block. Async-copy candidate keys to LDS (ASYNCcnt
// path), bitonic top-K sort, decode the K winner boxes (only reads of
// p_loc/anchors in the whole pipeline), WMMA-assisted IoU suppression mask,
// greedy NMS, outputs.
//
// WMMA use: union = (area_i + area_j) - inter. The pairwise area-sum over a
// 16x16 tile is the rank-2 product [area_i, 1] x [[1..1],[area_j..]], i.e.
// one V_WMMA_F32_16X16X4_F32 per tile. A-frag (16x4 f32): lanes 0-15 hold
// M=lane with VGPR0=K0, VGPR1=K1; lanes 16-31 hold K2/K3 (zero here).
// B-frag (4x16): lanes 0-15 hold N=lane rows K0/K1; hi lanes K2/K3 (zero).
// ---------------------------------------------------------------------------
__global__ __launch_bounds__(1024)
void topk_nms_kernel(const float* __restrict__ loc,
                     const float* __restrict__ anchors,
                     const int* __restrict__ counts,
                     const unsigned long long* __restrict__ keys,
                     float* __restrict__ out) {
  __shared__ unsigned long long skey[SORT_N];    // 64 KB
  __shared__ float bl[PAD], bt[PAD], br[PAD], bb[PAD];
  __shared__ float sarea[PAD];
  __shared__ float sscore[PAD];
  __shared__ int   svalid[PAD];
  __shared__ unsigned int sup[K_][10];           // 12 KB suppression bitmask
  __shared__ unsigned int keepw[10];

  const int b = blockIdx.x;
  int n = counts[b]; if (n > CAP) n = CAP;
  const unsigned long long* kb = keys + (size_t)b * CAP;

  // Stage keys into LDS with the CDNA5 async copy (tracked by ASYNCcnt);
  // pad the rest with 0 (sorts below every survivor: score bits >= 0x3F000000).
  for (int t = threadIdx.x; t < SORT_N; t += blockDim.x) {
    if (t < n) {
      const unsigned lds_addr = (unsigned)(unsigned long long)
          (__attribute__((address_space(3))) unsigned long long*)&skey[t];
      const unsigned long long gaddr = (unsigned long long)(size_t)(kb + t);
      asm volatile("global_load_async_to_lds_b64 %0, %1, off"
                   :: "v"(lds_addr), "v"(gaddr) : "memory");
    } else {
      skey[t] = 0ull;
    }
  }
  asm volatile("s_wait_asynccnt 0x0" ::: "memory");  // own-wave async complete
  __syncthreads();                                   // cross-wave visibility

  // Bitonic sort, descending.
  for (int k = 2; k <= SORT_N; k <<= 1) {
    for (int j = k >> 1; j > 0; j >>= 1) {
      for (int t = threadIdx.x; t < (SORT_N >> 1); t += blockDim.x) {
        const int i = ((t & ~(j - 1)) << 1) | (t & (j - 1));
        const int p = i | j;
        const bool desc = (i & k) == 0;
        const unsigned long long a = skey[i], c = skey[p];
        const bool sw = desc ? (a < c) : (a > c);
        if (sw) { skey[i] = c; skey[p] = a; }
      }
      __syncthreads();
    }
  }

  // Decode top-K boxes; pad rows [K_, PAD) with degenerate zero boxes.
  for (int t = threadIdx.x; t < PAD; t += blockDim.x) {
    float score = 0.0f; int valid = 0;
    float l = 0.f, tp = 0.f, r = 0.f, bo = 0.f;
    if (t < K_) {
      const unsigned long long key = skey[t];
      score = __uint_as_float((unsigned int)(key >> 32));
      valid = (score >= 0.5f);
      if (valid) {
        const int idx = (int)(~(unsigned int)key);
        const float4 lc = *(const float4*)(loc + ((size_t)b * A_ + (size_t)idx) * 4);
        const float4 an = *(const float4*)(anchors + (size_t)idx * 4);
        const float cx = an.x + lc.x * 0.1f * an.z;     // VAR0
        const float cy = an.y + lc.y * 0.1f * an.w;
        const float w  = an.z * __expf(lc.z * 0.2f);    // VAR1
        const float h  = an.w * __expf(lc.w * 0.2f);
        l = cx - 0.5f * w; tp = cy - 0.5f * h;
        r = cx + 0.5f * w; bo = cy + 0.5f * h;
      }
    }
    bl[t] = l; bt[t] = tp; br[t] = r; bb[t] = bo;
    sarea[t]  = fmaxf(r - l, 0.f) * fmaxf(bo - tp, 0.f);
    sscore[t] = score; svalid[t] = valid;
  }
  for (int t = threadIdx.x; t < K_ * 10; t += blockDim.x)
    ((unsigned int*)sup)[t] = 0u;
  __syncthreads();

  if (threadIdx.x < 10) {
    unsigned int w = 0;
    for (int bit = 0; bit < 32; ++bit) {
      const int s = (int)threadIdx.x * 32 + bit;
      if (s < K_ && svalid[s]) w |= (1u << bit);
    }
    keepw[threadIdx.x] = w;                        // keep init = valid
  }

  // IoU tiles: pairwise area-sum via WMMA, intersection via VALU.
  // Division-free threshold: inter/max(u,1e-9) > θ  <=>  inter > θ*max(u,1e-9).
  const int wave = threadIdx.x >> 5;
  const int lane = threadIdx.x & 31;
  const int nwaves = blockDim.x >> 5;
  for (int tile = wave; tile < NT * NT; tile += nwaves) {
    const int ti = tile / NT, tj = tile % NT;
    const int i0 = ti * 16, j0 = tj * 16;
    v2f af, bf; v8f c = {};
    const bool lo = (lane < 16);
    af.x = lo ? sarea[i0 + lane] : 0.0f;   // A[M=lane][K=0] = area_i
    af.y = lo ? 1.0f             : 0.0f;   // A[M=lane][K=1] = 1
    bf.x = lo ? 1.0f             : 0.0f;   // B[K=0][N=lane] = 1
    bf.y = lo ? sarea[j0 + lane] : 0.0f;   // B[K=1][N=lane] = area_j
    const v8f d = __builtin_amdgcn_wmma_f32_16x16x4_f32(
        false, af, false, bf, (short)0, c, false, false);
    const int colL  = lane & 15;
    const int rbase = lo ? 0 : 8;
    const int col   = j0 + colL;
    const float cl = bl[col], ct = bt[col], cr = br[col], cbm = bb[col];
#pragma unroll
    for (int v = 0; v < 8; ++v) {
      const int row = i0 + rbase + v;      // C/D layout: VGPR v -> M=v (+8 hi)
      const float pairsum = d[v];          // area_i + area_j
      const float il = fmaxf(bl[row], cl);
      const float it = fmaxf(bt[row], ct);
      const float ir = fminf(br[row], cr);
      const float ib = fminf(bb[row], cbm);
      const float inter = fmaxf(ir - il, 0.f) * fmaxf(ib - it, 0.f);
      const float uni = fmaxf(pairsum - inter, 1e-9f);
      if (row < K_ && col < K_ && col > row && inter > THR_NMS * uni)
        atomicOr(&sup[row][col >> 5], 1u << (col & 31));
    }
  }
  __syncthreads();

  // Greedy NMS: keep-words cached in registers (statically indexed), LDS
  // reads only for rows that are actually kept.
  if (threadIdx.x == 0) {
    unsigned int kw0 = keepw[0], kw1 = keepw[1], kw2 = keepw[2], kw3 = keepw[3];
    unsigned int kw4 = keepw[4], kw5 = keepw[5], kw6 = keepw[6], kw7 = keepw[7];
    unsigned int kw8 = keepw[8], kw9 = keepw[9];
    unsigned int kw[10];
#define KW_LOAD()  { kw[0]=kw0;kw[1]=kw1;kw[2]=kw2;kw[3]=kw3;kw[4]=kw4; \
                     kw[5]=kw5;kw[6]=kw6;kw[7]=kw7;kw[8]=kw8;kw[9]=kw9; }
#define KW_STORE() { kw0=kw[0];kw1=kw[1];kw2=kw[2];kw3=kw[3];kw4=kw[4]; \
                     kw5=kw[5];kw6=kw[6];kw7=kw[7];kw8=kw[8];kw9=kw[9]; }
#pragma unroll
    for (int w = 0; w < 10; ++w) {
      KW_LOAD();
      const int nbits = (w == 9) ? (K_ - 9 * 32) : 32;
      for (int bit = 0; bit < nbits; ++bit) {
        const int i = w * 32 + bit;
        if (kw[w] & (1u << bit)) {
#pragma unroll
          for (int w2 = 0; w2 < 10; ++w2) kw[w2] &= ~sup[i][w2];
        }
      }
      KW_STORE();
    }
    KW_LOAD();
#pragma unroll
    for (int w = 0; w < 10; ++w) keepw[w] = kw[w];
  }
  __syncthreads();

  // Outputs, flat concat in return order: ids | boxes*4 | labels | scores | keep
  const int BK = B_ * K_;
  for (int t = threadIdx.x; t < K_; t += blockDim.x) {
    const int slot = b * K_ + t;
    const int kp = (keepw[t >> 5] >> (t & 31)) & 1;
    const float kf = (float)kp;
    out[slot] = kp ? (float)b : -1.0f;                 // ids_batch2
    float* bx = out + BK + (size_t)slot * 4;           // p_boxes_ltrb2
    bx[0] = bl[t] * kf; bx[1] = bt[t] * kf;
    bx[2] = br[t] * kf; bx[3] = bb[t] * kf;
    out[5 * BK + slot] = kf;                           // p_labels2
    out[6 * BK + slot] = sscore[t] * kf;               // p_scores2
    out[7 * BK + slot] = kf;                           // keep
  }
}

// ---------------------------------------------------------------------------
extern "C" void kernel_launch(void* const* d_in, const int* in_sizes, int n_in,
                              void* d_out, int out_size, void* d_ws, size_t ws_size,
                              hipStream_t stream) {
  const float* p_loc   = (const float*)d_in[0];  // [B,A,4]
  const float* p_conf  = (const float*)d_in[1];  // [B,A,1]
  // d_in[2] = p_landms: dead compute in the reference; never read.
  const float* anchors = (const float*)d_in[3];  // [A,4]

  int* counts = (int*)d_ws;
  unsigned long long* keys =
      (unsigned long long*)((char*)d_ws + 128);  // B*CAP u64 = 1 MB

  reset_counts_kernel<<<1, 32, 0, stream>>>(counts);
  scan_conf_kernel<<<B_ * BLOCKS_PER_BATCH, 256, 0, stream>>>(p_conf, counts, keys);
  topk_nms_kernel<<<B_, 1024, 0, stream>>>(p_loc, anchors, counts, keys,
                                           (float*)d_out);
}